// NeuralHMM_8186207666599
// MI455X (gfx1250) — compile-verified
//
#include <hip/hip_runtime.h>
#include <math.h>

// Problem dims (fixed by the reference)
#define Bsz 64
#define Tn  128
#define Sn  8
#define Hn  128
#define Vv  30000
#define Dn  768

typedef __attribute__((ext_vector_type(2))) float v2f;
typedef __attribute__((ext_vector_type(8))) float v8f;

// ---------------- block reductions (LDS tree) ----------------
__device__ __forceinline__ float block_reduce_max(float v, float* red, int tid, int n) {
    red[tid] = v; __syncthreads();
    for (int off = n >> 1; off > 0; off >>= 1) {
        if (tid < off) red[tid] = fmaxf(red[tid], red[tid + off]);
        __syncthreads();
    }
    float r = red[0]; __syncthreads();
    return r;
}
__device__ __forceinline__ float block_reduce_sum(float v, float* red, int tid, int n) {
    red[tid] = v; __syncthreads();
    for (int off = n >> 1; off > 0; off >>= 1) {
        if (tid < off) red[tid] += red[tid + off];
        __syncthreads();
    }
    float r = red[0]; __syncthreads();
    return r;
}

// ---------------- 1) lse[s,h] = logsumexp_v emis[s,h,:] ----------------
__global__ __launch_bounds__(256) void emis_lse_kernel(const float* __restrict__ emis,
                                                       float* __restrict__ lse) {
    __shared__ float red[256];
    int row = blockIdx.x;              // row = s*Hn + h, 1024 rows
    int tid = threadIdx.x;
    const float* p = emis + (size_t)row * Vv;
    float m = -INFINITY;
    for (int v = tid; v < Vv; v += 256) m = fmaxf(m, p[v]);
    m = block_reduce_max(m, red, tid, 256);
    float s = 0.f;
    for (int v = tid; v < Vv; v += 256) s += expf(p[v] - m);
    s = block_reduce_sum(s, red, tid, 256);
    if (tid == 0) lse[row] = m + logf(s);
}

// ---------------- 2) E[bt,h] = mean_s (emis[s,h,obs[bt,s]] - lse[s,h]) ----------------
__global__ __launch_bounds__(128) void e_gather_kernel(const float* __restrict__ emis,
                                                       const float* __restrict__ lse,
                                                       const int* __restrict__ obs,
                                                       float* __restrict__ E) {
    __shared__ int sv[Sn];
    int bt = blockIdx.x;               // 8192 blocks
    int h  = threadIdx.x;              // 128 threads
    if (h < Sn) sv[h] = obs[bt * Sn + h];
    __syncthreads();
    float acc = 0.f;
#pragma unroll
    for (int s = 0; s < Sn; ++s) {
        int v = sv[s];
        acc += emis[((size_t)s * Hn + h) * Vv + v] - lse[s * Hn + h];
    }
    E[(size_t)bt * Hn + h] = acc * (1.0f / (float)Sn);
}

// ---------------- 3) Hh = gelu(embs @ W^T + b), WMMA f32 16x16x4 ----------------
#define A_STRIDE 772  // 768 + 4 pad: conflict-free ds reads down the M dimension
__global__ __launch_bounds__(256) void emb2h_gemm_gelu_kernel(const float* __restrict__ embs,
                                                              const float* __restrict__ W,   // [Hn, Dn]
                                                              const float* __restrict__ bias,
                                                              float* __restrict__ Hh) {
    __shared__ float sA[16 * A_STRIDE];  // 16 x 768 A tile (f32)
    int tid  = threadIdx.x;
    int row0 = blockIdx.x * 16;          // 512 blocks cover M = 8192
    // cooperative, fully coalesced tile load (tile is 12288 contiguous floats)
    for (int r = 0; r < 16; ++r)
        for (int c = tid; c < Dn; c += 256)
            sA[r * A_STRIDE + c] = embs[(size_t)(row0 + r) * Dn + c];
    __syncthreads();

    int lane = tid & 31;
    int wv   = tid >> 5;                 // 8 waves, each owns a 16-col N tile
    int l16  = lane & 15;
    int hi   = lane >> 4;                // lanes 16-31 carry K+2, K+3
    int col  = wv * 16 + l16;            // N index (0..127)

    const float* wrow = W  + (size_t)col * Dn + 2 * hi;  // B[k,n] = W[n,k]
    const float* arow = sA + l16 * A_STRIDE + 2 * hi;

    v8f acc = {};
    for (int k = 0; k < Dn; k += 4) {
        v2f a;  a.x = arow[k]; a.y = arow[k + 1];        // A frag: 16x4 f32 layout
        v2f bb; bb.x = wrow[k]; bb.y = wrow[k + 1];      // B frag: 4x16 f32 layout
        acc = __builtin_amdgcn_wmma_f32_16x16x4_f32(
            /*neg_a=*/false, a, /*neg_b=*/false, bb,
            /*c_mod=*/(short)0, acc, /*reuse_a=*/false, /*reuse_b=*/false);
    }

    float bi = bias[col];
#pragma unroll
    for (int i = 0; i < 8; ++i) {        // C/D layout: VGPR i -> M = i + 8*hi
        int row = row0 + i + 8 * hi;
        float x = acc[i] + bi;
        float g = 0.5f * x * (1.0f + erff(x * 0.70710678118654752f)); // exact gelu
        Hh[(size_t)row * Hn + col] = g;
    }
}

// ---------------- 4) score[bt] = sigmoid(Hh[bt,:] . w + b) ----------------
__global__ __launch_bounds__(256) void score_kernel(const float* __restrict__ Hh,
                                                    const float* __restrict__ w,
                                                    const float* __restrict__ b,
                                                    float* __restrict__ score) {
    int gw   = blockIdx.x * 8 + (threadIdx.x >> 5);  // one wave32 per bt; 1024 blocks
    int lane = threadIdx.x & 31;
    const float* hrow = Hh + (size_t)gw * Hn;
    float d = 0.f;
    for (int i = lane; i < Hn; i += 32) d += hrow[i] * w[i];
    for (int off = 16; off > 0; off >>= 1) d += __shfl_down(d, off, 32);
    if (lane == 0) score[gw] = 1.0f / (1.0f + expf(-(d + b[0])));
}

// ---------------- 5) tran = softmax(unnormalized_tran, axis=-1) ----------------
__global__ __launch_bounds__(128) void tran_softmax_kernel(const float* __restrict__ U,
                                                           float* __restrict__ tran) {
    __shared__ float red[128];
    int row = blockIdx.x, j = threadIdx.x;
    float x = U[row * Hn + j];
    float m = block_reduce_max(x, red, j, 128);
    float e = expf(x - m);
    float s = block_reduce_sum(e, red, j, 128);
    tran[row * Hn + j] = e / s;
}

// ---------------- 6) sequential HMM forward scan; one block per batch ----------------
// new[b,j] = E[b,t,j] + m + log( Σ_k w1[k]·exp(cw[k]·h[j] − mk[k]) + (1−s)·Σ_k p[k]·tran[k,j] )
// with m = max_k la, p = exp(la−m), w1[k] = p[k]·s/denom[k], mk/denom = stable softmax pieces.
// conv_b cancels inside softmax over j, so it is (correctly) ignored.
__global__ __launch_bounds__(128) void hmm_scan_kernel(const float* __restrict__ E,
                                                       const float* __restrict__ Hh,
                                                       const float* __restrict__ score,
                                                       const float* __restrict__ tran,
                                                       const float* __restrict__ conv_w,
                                                       const int* __restrict__ lengths,
                                                       float* __restrict__ out) {
    __shared__ float red[128];
    __shared__ float s_h[128], s_cw[128], s_mk[128], s_w1[128], s_pk[128];
    int b = blockIdx.x, j = threadIdx.x;
    s_cw[j] = conv_w[j];
    int last = lengths[b] - 1;

    // log priors: full(0.1), [0]=1.0, log_softmax (closed form)
    float lprior = ((j == 0) ? 1.0f : 0.1f) - logf(expf(1.0f) + 127.0f * expf(0.1f));
    float la = E[(size_t)(b * Tn) * Hn + j] + lprior;
    {
        float m2 = block_reduce_max(la, red, j, 128);
        float sm = block_reduce_sum(expf(la - m2), red, j, 128);
        if (j == 0 && last == 0) out[b] = m2 + logf(sm);
    }

    for (int t = 1; t < Tn; ++t) {
        int bt = b * Tn + t;
        float hj = Hh[(size_t)bt * Hn + j];
        float sc = score[bt];
        s_h[j] = hj;

        float m = block_reduce_max(la, red, j, 128);          // barrier: s_h visible
        float pk = expf(la - m);
        s_pk[j] = pk;
        float hmax = block_reduce_max(hj, red, j, 128);
        float hmin = -block_reduce_max(-hj, red, j, 128);

        // thread j plays role of source state k: stable softmax denom of cw[k]*h[:]
        float cw = s_cw[j];
        float mk = (cw >= 0.f) ? cw * hmax : cw * hmin;
        float denom = 0.f;
        for (int jj = 0; jj < 128; ++jj) denom += expf(cw * s_h[jj] - mk);
        s_mk[j] = mk;
        s_w1[j] = pk * sc / denom;
        __syncthreads();

        // thread j plays role of destination state: accumulate over k
        float acc1 = 0.f, acc2 = 0.f;
        for (int k = 0; k < 128; ++k) {
            acc1 += s_w1[k] * expf(s_cw[k] * hj - s_mk[k]);
            acc2 += s_pk[k] * tran[k * Hn + j];               // coalesced across j
        }
        float tot = acc1 + (1.0f - sc) * acc2;
        la = E[(size_t)bt * Hn + j] + m + logf(tot);

        float m2 = block_reduce_max(la, red, j, 128);
        float sm = block_reduce_sum(expf(la - m2), red, j, 128);
        if (j == 0 && t == last) out[b] = m2 + logf(sm);
        __syncthreads();
    }
}

// ---------------- host launcher ----------------
extern "C" void kernel_launch(void* const* d_in, const int* in_sizes, int n_in,
                              void* d_out, int out_size, void* d_ws, size_t ws_size,
                              hipStream_t stream) {
    (void)in_sizes; (void)n_in; (void)out_size; (void)ws_size;
    const float* embs    = (const float*)d_in[0];   // [B,T,D]
    const int*   obs     = (const int*)  d_in[1];   // [B,T,S]
    const int*   lengths = (const int*)  d_in[2];   // [B]
    const float* emis    = (const float*)d_in[3];   // [S,H,V]
    const float* utran   = (const float*)d_in[4];   // [H,H]
    const float* e2h_w   = (const float*)d_in[5];   // [H,D]
    const float* e2h_b   = (const float*)d_in[6];   // [H]
    const float* h2s_w   = (const float*)d_in[7];   // [1,H]
    const float* h2s_b   = (const float*)d_in[8];   // [1]
    const float* conv_w  = (const float*)d_in[9];   // [H]
    // d_in[10] = conv_b: cancels inside softmax over j -> unused.
    float* out = (float*)d_out;                     // [B,1] f32

    float* ws    = (float*)d_ws;
    float* lse   = ws;                              // 1024
    float* E     = lse + (Sn * Hn);                 // 8192*128
    float* Hh    = E + (size_t)Bsz * Tn * Hn;       // 8192*128
    float* score = Hh + (size_t)Bsz * Tn * Hn;      // 8192
    float* tran  = score + Bsz * Tn;                // 128*128

    emis_lse_kernel      <<<Sn * Hn,        256, 0, stream>>>(emis, lse);
    e_gather_kernel      <<<Bsz * Tn,       128, 0, stream>>>(emis, lse, obs, E);
    emb2h_gemm_gelu_kernel<<<(Bsz * Tn) / 16, 256, 0, stream>>>(embs, e2h_w, e2h_b, Hh);
    score_kernel         <<<(Bsz * Tn) / 8, 256, 0, stream>>>(Hh, h2s_w, h2s_b, score);
    tran_softmax_kernel  <<<Hn,             128, 0, stream>>>(utran, tran);
    hmm_scan_kernel      <<<Bsz,            128, 0, stream>>>(E, Hh, score, tran, conv_w, lengths, out);
}